// VQClassifier_26405458936340
// MI455X (gfx1250) — compile-verified
//
#include <hip/hip_runtime.h>
#include <hip/hip_bf16.h>

// Problem constants from the reference.
#define B_   32
#define T_   1024
#define D_   1024   // KEY_DIM
#define NE_  1024   // number of codebook entries
#define ED_  1024   // embedding dim
#define EPS_ 1e-12f

typedef float v2f __attribute__((ext_vector_type(2)));
typedef float v8f __attribute__((ext_vector_type(8)));

// ---------------------------------------------------------------------------
// K1: row-normalize cls_weight -> clnorm   (x / max(||x||, EPS), matches ref)
// ---------------------------------------------------------------------------
__global__ void k_normalize_rows(const float* __restrict__ w, float* __restrict__ out) {
    __shared__ float red[256];
    const int row = blockIdx.x;
    const int tid = threadIdx.x;
    const float* src = w + (size_t)row * D_;
    float ss = 0.f;
    for (int k = tid; k < D_; k += 256) { const float v = src[k]; ss = fmaf(v, v, ss); }
    red[tid] = ss;
    __syncthreads();
    for (int s = 128; s > 0; s >>= 1) {
        if (tid < s) red[tid] += red[tid + s];
        __syncthreads();
    }
    const float scale = 1.0f / fmaxf(sqrtf(red[0]), EPS_);
    float* dst = out + (size_t)row * D_;
    for (int k = tid; k < D_; k += 256) dst[k] = src[k] * scale;
}

// ---------------------------------------------------------------------------
// K2: t=0 score rows via exact-f32 WMMA:  S0[32,1024] = KS[:,0,:] . clnorm^T
//     One wave per 16x16 tile; K-loop of V_WMMA_F32_16X16X4_F32.
//     Skipping key_soft normalization is exact: it scales each score row by a
//     positive scalar, which cannot change the argmax.
// ---------------------------------------------------------------------------
__global__ void k_score_t0_wmma(const float* __restrict__ ks,
                                const float* __restrict__ clnorm,
                                float* __restrict__ s0) {
    const int lane = threadIdx.x;        // 32 threads = 1 wave (wave32)
    const int lo = lane & 15;
    const int hi = lane >> 4;
    // A-matrix: row m = batch b at t=0; 16x4 f32 frag = float2 per lane:
    //   lanes 0-15 hold K = k0+{0,1}; lanes 16-31 hold K = k0+{2,3}.
    const int mg = blockIdx.y * 16 + lo;
    const float* arow = ks + (size_t)mg * ((size_t)T_ * D_);   // t = 0 row
    // B-matrix: 4x16 (KxN); B[k][n] = clnorm[n][k] -> contiguous float2 in k.
    const int ng = blockIdx.x * 16 + lo;
    const float* brow = clnorm + (size_t)ng * D_;

    v8f acc = {};
    #pragma unroll 8
    for (int k = 0; k < D_; k += 4) {
        const v2f a = *(const v2f*)(arow + k + 2 * hi);
        const v2f b = *(const v2f*)(brow + k + 2 * hi);
        acc = __builtin_amdgcn_wmma_f32_16x16x4_f32(
                  false, a, false, b, (short)0, acc, false, false);
    }
    // C/D layout: VGPR r -> M = r + 8*hi, N = lane&15 (per 16x16 f32 layout).
    const int col = blockIdx.x * 16 + lo;
    #pragma unroll
    for (int r = 0; r < 8; ++r) {
        const int m = blockIdx.y * 16 + r + 8 * hi;
        s0[(size_t)m * NE_ + col] = acc[r];
    }
}

// ---------------------------------------------------------------------------
// K2b: per-batch argmax over the t=0 score row (first-max on ties, like jnp).
// ---------------------------------------------------------------------------
__global__ void k_argmax_t0(const float* __restrict__ s0, int* __restrict__ enc) {
    __shared__ float sv[256];
    __shared__ int   si[256];
    const int b = blockIdx.x;
    const int tid = threadIdx.x;
    const float* row = s0 + (size_t)b * NE_;
    float bv = -3.402823466e38f;
    int   bi = 0;
    #pragma unroll
    for (int j = 0; j < NE_ / 256; ++j) {
        const int c = tid * (NE_ / 256) + j;       // ascending -> strict > keeps first max
        const float v = row[c];
        if (v > bv) { bv = v; bi = c; }
    }
    sv[tid] = bv; si[tid] = bi;
    __syncthreads();
    for (int s = 128; s > 0; s >>= 1) {
        if (tid < s) {
            if (sv[tid + s] > sv[tid] ||
                (sv[tid + s] == sv[tid] && si[tid + s] < si[tid])) {
                sv[tid] = sv[tid + s]; si[tid] = si[tid + s];
            }
        }
        __syncthreads();
    }
    if (tid == 0) enc[(size_t)b * T_] = si[0];
}

// ---------------------------------------------------------------------------
// K3: greedy stay-or-advance chain. One wave32 per batch. Each step fuses the
//     two needed dot products (cols ind, ind+1); a butterfly reduce leaves the
//     totals in every lane so all lanes update `ind` identically.
//     Speculative global_prefetch hides next-step L2 latency (index advances
//     by at most 1 per step, so clnorm row ind+2 is the only new row).
//     Comparisons are invariant to key_soft row normalization (positive scale).
// ---------------------------------------------------------------------------
__global__ void k_greedy_chain(const float* __restrict__ ks,
                               const float* __restrict__ clnorm,
                               int* __restrict__ enc) {
    const int b = blockIdx.x;
    const int l = threadIdx.x;                 // lane 0..31
    const float* krow_base = ks + (size_t)b * T_ * D_;
    int ind = enc[(size_t)b * T_];
    for (int t = 1; t < T_; ++t) {
        const float* krow = krow_base + (size_t)t * D_;
        const int ind1 = min(ind + 1, NE_ - 1);
        const float* ch = clnorm + (size_t)ind  * D_;
        const float* cn = clnorm + (size_t)ind1 * D_;
        if (t + 1 < T_) {
            // 32 lanes x 128B covers the whole next 4KB key_soft row.
            __builtin_prefetch(krow + D_ + l * 32, 0, 0);
            const int pn = min(ind + 2, NE_ - 1);
            __builtin_prefetch(clnorm + (size_t)pn * D_ + l * 32, 0, 0);
        }
        float dh = 0.f, dn = 0.f;
        #pragma unroll
        for (int j = 0; j < D_ / 32; ++j) {
            const int k = l + 32 * j;          // coalesced across the wave
            const float x = krow[k];
            dh = fmaf(x, ch[k], dh);
            dn = fmaf(x, cn[k], dn);
        }
        #pragma unroll
        for (int off = 16; off > 0; off >>= 1) {
            dh += __shfl_xor(dh, off, 32);
            dn += __shfl_xor(dn, off, 32);
        }
        ind = (dh >= dn) ? ind : ind1;
        if (l == 0) enc[(size_t)b * T_ + t] = ind;
    }
}

// ---------------------------------------------------------------------------
// K4: output = emb_weight[enc_idx]  (STE forward value). b128 row copies.
// ---------------------------------------------------------------------------
__global__ void k_gather_out(const int* __restrict__ enc,
                             const float* __restrict__ emb,
                             float* __restrict__ out) {
    const int r = blockIdx.x;                  // 0 .. B*T-1
    const int idx = enc[r];
    const float4* src = (const float4*)(emb + (size_t)idx * ED_);
    float4* dst = (float4*)(out + (size_t)r * ED_);
    dst[threadIdx.x] = src[threadIdx.x];       // 256 threads x float4 = 4KB row
}

// ---------------------------------------------------------------------------
extern "C" void kernel_launch(void* const* d_in, const int* in_sizes, int n_in,
                              void* d_out, int out_size, void* d_ws, size_t ws_size,
                              hipStream_t stream) {
    const float* key_soft = (const float*)d_in[0];   // [B, T, D] f32
    const float* cls_w    = (const float*)d_in[1];   // [NE, D]   f32
    const float* emb_w    = (const float*)d_in[2];   // [NE, ED]  f32
    float* out = (float*)d_out;                      // [B, T, ED] f32

    // Workspace layout (~4.3 MB total):
    float* clnorm = (float*)d_ws;                          // NE*D f32   (4 MB)
    float* s0     = clnorm + (size_t)NE_ * D_;             // B*NE f32   (128 KB)
    int*   enc    = (int*)(s0 + (size_t)B_ * NE_);         // B*T  int   (128 KB)

    k_normalize_rows<<<NE_, 256, 0, stream>>>(cls_w, clnorm);
    k_score_t0_wmma<<<dim3(NE_ / 16, B_ / 16), 32, 0, stream>>>(key_soft, clnorm, s0);
    k_argmax_t0<<<B_, 256, 0, stream>>>(s0, enc);
    k_greedy_chain<<<B_, 32, 0, stream>>>(key_soft, clnorm, enc);
    k_gather_out<<<B_ * T_, 256, 0, stream>>>(enc, emb_w, out);
}